// PostProcess_90065464197476
// MI455X (gfx1250) — compile-verified
//
#include <hip/hip_runtime.h>
#include <stdint.h>

#define MIN_THR 0.25f
#define NMS_THR 0.35f
#define NPAD    8192      // next pow2 >= N (N = 5000)
#define SORT_T  1024
#define NMS_TPB 64
#define MASK_W  128       // padded mask words per row (N <= 8192); 1KB per row
#define CHUNK   16        // rows prefetched per async chunk: 16KB = 32 x b128 x 32 lanes

// ---------------------------------------------------------------------------
// Kernel 1: one wave32 per query: sigmoid(max logit), argmax+1, box cxcywh->xyxy*scale
// ---------------------------------------------------------------------------
__global__ void score_box_kernel(const float* __restrict__ logits,
                                 const float* __restrict__ cxcywh,
                                 const int*   __restrict__ tsizes,
                                 float* __restrict__ scores,
                                 int*   __restrict__ labels,
                                 float* __restrict__ boxes,
                                 int Ntot, int Q, int C) {
  int lane = threadIdx.x & 31;
  int n = blockIdx.x * (blockDim.x >> 5) + (threadIdx.x >> 5);
  if (n >= Ntot) return;
  const float* row = logits + (size_t)n * C;
  float best = -INFINITY; int bidx = C;
  for (int c = lane; c < C; c += 32) {      // coalesced: lanes read consecutive classes
    float v = row[c];
    if (v > best) { best = v; bidx = c; }
  }
  for (int off = 16; off > 0; off >>= 1) {  // wave32 argmax reduce, tie -> smaller idx
    float ob = __shfl_down(best, off, 32);
    int   oi = __shfl_down(bidx, off, 32);
    if (ob > best || (ob == best && oi < bidx)) { best = ob; bidx = oi; }
  }
  if (lane == 0) {
    scores[n] = 1.0f / (1.0f + expf(-best));
    labels[n] = bidx + 1;
    int b = n / Q;
    float ih = (float)tsizes[2 * b + 0];
    float iw = (float)tsizes[2 * b + 1];
    float cx = cxcywh[(size_t)n * 4 + 0], cy = cxcywh[(size_t)n * 4 + 1];
    float w  = cxcywh[(size_t)n * 4 + 2], h  = cxcywh[(size_t)n * 4 + 3];
    boxes[(size_t)n * 4 + 0] = (cx - 0.5f * w) * iw;
    boxes[(size_t)n * 4 + 1] = (cy - 0.5f * h) * ih;
    boxes[(size_t)n * 4 + 2] = (cx + 0.5f * w) * iw;
    boxes[(size_t)n * 4 + 3] = (cy + 0.5f * h) * ih;
  }
}

// ---------------------------------------------------------------------------
// Kernel 2: maxcoord = max(all main-box coords) + 1.0  (single block reduce)
// ---------------------------------------------------------------------------
__global__ void maxcoord_kernel(const float* __restrict__ boxes, int n4,
                                float* __restrict__ outmax) {
  __shared__ float red[256];
  float m = -INFINITY;
  for (int i = threadIdx.x; i < n4; i += blockDim.x) m = fmaxf(m, boxes[i]);
  red[threadIdx.x] = m;
  __syncthreads();
  for (int s = blockDim.x >> 1; s > 0; s >>= 1) {
    if (threadIdx.x < (unsigned)s) red[threadIdx.x] = fmaxf(red[threadIdx.x], red[threadIdx.x + s]);
    __syncthreads();
  }
  if (threadIdx.x == 0) outmax[0] = red[0] + 1.0f;
}

// ---------------------------------------------------------------------------
// Kernel 3: full bitonic sort of N scores in one workgroup's LDS (64 KB of the
// WGP's 320 KB). u64 key = (descending-score-orderable << 32) | idx
// -> exactly stable argsort(-scores).
// ---------------------------------------------------------------------------
__global__ void __launch_bounds__(SORT_T)
sort_kernel(const float* __restrict__ scores, int N,
            int* __restrict__ sidx, float* __restrict__ sscore) {
  __shared__ unsigned long long key[NPAD];
  for (int i = threadIdx.x; i < NPAD; i += blockDim.x) {
    unsigned int k32;
    if (i < N) {
      unsigned int bits = __float_as_uint(scores[i]);
      unsigned int ord  = (bits & 0x80000000u) ? ~bits : (bits | 0x80000000u);
      k32 = ~ord;                         // ascending k32 == descending score
    } else {
      k32 = 0xFFFFFFFFu;                  // pads sort last
    }
    key[i] = ((unsigned long long)k32 << 32) | (unsigned int)i;
  }
  __syncthreads();
  for (unsigned int k = 2; k <= NPAD; k <<= 1) {
    for (unsigned int j = k >> 1; j > 0; j >>= 1) {
      for (unsigned int i = threadIdx.x; i < NPAD; i += blockDim.x) {
        unsigned int ixj = i ^ j;
        if (ixj > i) {
          bool up = ((i & k) == 0);
          unsigned long long a = key[i], b = key[ixj];
          if (up ? (a > b) : (a < b)) { key[i] = b; key[ixj] = a; }
        }
      }
      __syncthreads();
    }
  }
  for (int i = threadIdx.x; i < N; i += blockDim.x) {
    int idx = (int)(key[i] & 0xFFFFFFFFu);
    sidx[i]   = idx;
    sscore[i] = scores[idx];
  }
}

// ---------------------------------------------------------------------------
// Kernel 4: offset boxes in sorted order: box[order[i]] + label*maxcoord
// ---------------------------------------------------------------------------
__global__ void offbox_kernel(const int* __restrict__ sidx,
                              const float* __restrict__ boxes,
                              const int* __restrict__ labels,
                              const float* __restrict__ maxc,
                              int N, float* __restrict__ offb) {
  int i = blockIdx.x * blockDim.x + threadIdx.x;
  if (i >= N) return;
  int flat = sidx[i];
  float off = (float)labels[flat] * maxc[0];
  float4 b = ((const float4*)boxes)[flat];
  float4 o = { b.x + off, b.y + off, b.z + off, b.w + off };
  ((float4*)offb)[i] = o;
}

// ---------------------------------------------------------------------------
// Kernel 5: suppression bitmask, 64 rows x 64 cols per block. Column tile
// staged into LDS via CDNA5 async global->LDS b128 (ASYNCcnt) + barrier.
// Rows are written with a padded stride of MASK_W words so the scan kernel
// can stream whole chunks with compile-time-constant async-copy counts.
// ---------------------------------------------------------------------------
__global__ void iou_mask_kernel(const float* __restrict__ offb, int N, int mwords,
                                unsigned long long* __restrict__ mask) {
  __shared__ float colb[NMS_TPB * 4];
  int rowStart = blockIdx.y * NMS_TPB;
  int colStart = blockIdx.x * NMS_TPB;
  int i = rowStart + threadIdx.x;

  if (colStart + NMS_TPB - 1 <= rowStart) {   // no (j > i) bit possible
    if (i < N) mask[(size_t)i * MASK_W + blockIdx.x] = 0ull;
    return;
  }

  int col0 = colStart + threadIdx.x;
  if (col0 < N) {
    unsigned int ldsOff = (unsigned int)(uintptr_t)(&colb[threadIdx.x * 4]);
    unsigned long long ga = (unsigned long long)(uintptr_t)(offb + (size_t)col0 * 4);
    asm volatile("global_load_async_to_lds_b128 %0, %1, off"
                 : : "v"(ldsOff), "v"(ga) : "memory");
  } else {
    colb[threadIdx.x * 4 + 0] = 0.0f;
    colb[threadIdx.x * 4 + 1] = 0.0f;
    colb[threadIdx.x * 4 + 2] = 0.0f;
    colb[threadIdx.x * 4 + 3] = 0.0f;
  }
  asm volatile("s_wait_asynccnt 0" ::: "memory");
  __syncthreads();

  if (i < N) {
    float4 rb4 = ((const float4*)offb)[i];
    float rx1 = rb4.x, ry1 = rb4.y, rx2 = rb4.z, ry2 = rb4.w;
    float area_r = (rx2 - rx1) * (ry2 - ry1);
    unsigned long long word = 0ull;
    #pragma unroll 4
    for (int c = 0; c < NMS_TPB; ++c) {
      int j = colStart + c;
      if (j > i && j < N) {
        float cx1 = colb[c * 4 + 0], cy1 = colb[c * 4 + 1];
        float cx2 = colb[c * 4 + 2], cy2 = colb[c * 4 + 3];
        float area_c = (cx2 - cx1) * (cy2 - cy1);
        float ltx = fmaxf(rx1, cx1), lty = fmaxf(ry1, cy1);
        float rbx = fminf(rx2, cx2), rby = fminf(ry2, cy2);
        float w = fmaxf(rbx - ltx, 0.0f), h = fmaxf(rby - lty, 0.0f);
        float inter = w * h;
        float iou = inter / (area_r + area_c - inter + 1e-6f);
        if (iou > NMS_THR) word |= (1ull << c);
      }
    }
    mask[(size_t)i * MASK_W + blockIdx.x] = word;
  }
  (void)mwords;
}

// ---------------------------------------------------------------------------
// Kernel 6: single-wave greedy NMS + top-k + gather. Barrier-free (wave32
// lockstep). remv bitmask lives in VGPRs (lane owns words lane+32s). Mask rows
// are double-buffer prefetched into LDS with async b128 copies: each CHUNK=16
// rows is contiguous 16KB = exactly 32 async instructions, so
// s_wait_asynccnt <= 32 (in-order completion) overlaps the next chunk's copy
// with the current chunk's serial scan.
// ---------------------------------------------------------------------------
__device__ __forceinline__ void issue_chunk_async(unsigned long long gbase,
                                                  unsigned ldsBase, int lane) {
  #pragma unroll
  for (int t = 0; t < 32; ++t) {
    unsigned lo = ldsBase + (unsigned)(t * 512 + lane * 16);
    unsigned long long ga = gbase + (unsigned long long)(t * 512 + lane * 16);
    asm volatile("global_load_async_to_lds_b128 %0, %1, off"
                 : : "v"(lo), "v"(ga) : "memory");
  }
}

__global__ void __launch_bounds__(32)
nms_finalize_kernel(const unsigned long long* __restrict__ mask,
                    const float* __restrict__ sscore,
                    const int*   __restrict__ sidx,
                    const float* __restrict__ boxes,
                    const int*   __restrict__ labels,
                    const float* __restrict__ pscores,
                    const int*   __restrict__ plabels,
                    const float* __restrict__ pboxes,
                    int N, int mwords, int K,
                    float* __restrict__ out) {
  __shared__ unsigned long long buf[2][CHUNK][MASK_W];   // 2 x 16KB double buffer
  __shared__ int topIdx[1024];
  int lane = threadIdx.x;

  unsigned long long r0 = 0, r1 = 0, r2 = 0, r3 = 0;     // remv words lane+0/32/64/96
  int cnt = 0;
  int nchunks = (N + CHUNK - 1) / CHUNK;

  unsigned lb0 = (unsigned)(uintptr_t)(&buf[0][0][0]);
  unsigned lb1 = (unsigned)(uintptr_t)(&buf[1][0][0]);
  unsigned long long gbase = (unsigned long long)(uintptr_t)mask;

  issue_chunk_async(gbase, lb0, lane);                   // prefetch chunk 0

  for (int c = 0; c < nchunks; ++c) {
    int row0 = c * CHUNK;
    if (c + 1 < nchunks) {                               // prefetch next chunk
      issue_chunk_async(gbase + (unsigned long long)(row0 + CHUNK) * (MASK_W * 8ull),
                        ((c + 1) & 1) ? lb1 : lb0, lane);
      asm volatile("s_wait_asynccnt 0x20" ::: "memory"); // oldest 32 (chunk c) done
    } else {
      asm volatile("s_wait_asynccnt 0x0" ::: "memory");
    }

    float myscore = -2.0f;                               // lane r holds score of row r
    if (row0 + lane < N) myscore = sscore[row0 + lane];

    unsigned long long (*bufp)[MASK_W] = buf[c & 1];
    int rows = N - row0; if (rows > CHUNK) rows = CHUNK;

    for (int r = 0; r < rows; ++r) {
      int i = row0 + r;
      int w = i >> 6, bpos = i & 63;
      int owner = w & 31, slot = w >> 5;
      unsigned long long my = (slot == 0) ? r0 : (slot == 1) ? r1 : (slot == 2) ? r2 : r3;
      unsigned shi = __shfl((unsigned)(my >> 32), owner, 32);
      unsigned slo = __shfl((unsigned)(my & 0xFFFFFFFFu), owner, 32);
      unsigned long long wv = ((unsigned long long)shi << 32) | slo;
      float sc = __shfl(myscore, r, 32);
      bool alive = (((wv >> bpos) & 1ull) == 0ull) && (sc >= MIN_THR); // uniform
      if (alive) {
        if (cnt < K) { if (lane == 0) topIdx[cnt] = i; cnt++; }
        if (lane < mwords)      r0 |= bufp[r][lane];
        if (lane + 32 < mwords) r1 |= bufp[r][lane + 32];
        if (lane + 64 < mwords) r2 |= bufp[r][lane + 64];
        if (lane + 96 < mwords) r3 |= bufp[r][lane + 96];
      }
    }
  }

  // Phase 2: pad top-k with non-kept positions ascending (-1 ties in reference).
  // Final remv bit i == remv bit i at step i (suppressors only set bits j > i).
  for (int i = 0; i < N && cnt < K; ++i) {
    int w = i >> 6, bpos = i & 63;
    int owner = w & 31, slot = w >> 5;
    unsigned long long my = (slot == 0) ? r0 : (slot == 1) ? r1 : (slot == 2) ? r2 : r3;
    unsigned shi = __shfl((unsigned)(my >> 32), owner, 32);
    unsigned slo = __shfl((unsigned)(my & 0xFFFFFFFFu), owner, 32);
    unsigned long long wv = ((unsigned long long)shi << 32) | slo;
    bool kept = (((wv >> bpos) & 1ull) == 0ull) && (sscore[i] >= MIN_THR);
    if (!kept) { if (lane == 0) topIdx[cnt] = i; cnt++; }
  }
  for (; cnt < K; ++cnt) { if (lane == 0) topIdx[cnt] = 0; }

  // Phase 3: gather outputs: det | mot | cls | boxes | pcls | pboxes (12*K floats)
  for (int t = lane; t < K; t += 32) {
    int pos  = topIdx[t];
    int flat = sidx[pos];
    float ds = sscore[pos];
    out[t]         = ds;
    out[K + t]     = pscores[flat] * ds;
    out[2 * K + t] = (float)labels[flat];
    float4 b = ((const float4*)boxes)[flat];
    out[3 * K + t * 4 + 0] = b.x;
    out[3 * K + t * 4 + 1] = b.y;
    out[3 * K + t * 4 + 2] = b.z;
    out[3 * K + t * 4 + 3] = b.w;
    out[7 * K + t] = (float)plabels[flat];
    float4 pb = ((const float4*)pboxes)[flat];
    out[8 * K + t * 4 + 0] = pb.x;
    out[8 * K + t * 4 + 1] = pb.y;
    out[8 * K + t * 4 + 2] = pb.z;
    out[8 * K + t * 4 + 3] = pb.w;
  }
}

// ---------------------------------------------------------------------------
extern "C" void kernel_launch(void* const* d_in, const int* in_sizes, int n_in,
                              void* d_out, int out_size, void* d_ws, size_t ws_size,
                              hipStream_t stream) {
  const float* pred_logits   = (const float*)d_in[0];
  const float* pred_boxes    = (const float*)d_in[1];
  const float* puppet_logits = (const float*)d_in[2];
  const float* puppet_boxes  = (const float*)d_in[3];
  const int*   tsizes        = (const int*)d_in[4];
  (void)n_in; (void)ws_size;

  int B = in_sizes[4] / 2;
  if (B < 1) B = 1;
  int Ntot = in_sizes[1] / 4;
  int Q    = Ntot / B;
  int C    = in_sizes[0] / Ntot;
  int N    = Ntot;                          // 5000
  int K    = out_size / 12;                 // topk (100)
  int mwords = (N + 63) / 64;               // 79
  int rowsAlloc = ((N + CHUNK - 1) / CHUNK) * CHUNK;   // chunk-padded mask rows

  char* p = (char*)d_ws;
  auto carve = [&](size_t bytes) -> void* {
    void* r = (void*)p;
    p += (bytes + 255) & ~(size_t)255;
    return r;
  };
  float* scores  = (float*)carve((size_t)N * 4);
  int*   labels  = (int*)  carve((size_t)N * 4);
  float* boxes   = (float*)carve((size_t)N * 16);
  float* pscores = (float*)carve((size_t)N * 4);
  int*   plabels = (int*)  carve((size_t)N * 4);
  float* pboxs   = (float*)carve((size_t)N * 16);
  int*   sidx    = (int*)  carve((size_t)N * 4);
  float* sscore  = (float*)carve((size_t)N * 4);
  float* maxc    = (float*)carve(16);
  float* offb    = (float*)carve((size_t)N * 16);
  unsigned long long* mask =
      (unsigned long long*)carve((size_t)rowsAlloc * MASK_W * 8);

  {
    int wavesPerBlock = 8;
    int blocks = (N + wavesPerBlock - 1) / wavesPerBlock;
    score_box_kernel<<<blocks, 256, 0, stream>>>(pred_logits, pred_boxes, tsizes,
                                                 scores, labels, boxes, N, Q, C);
    score_box_kernel<<<blocks, 256, 0, stream>>>(puppet_logits, puppet_boxes, tsizes,
                                                 pscores, plabels, pboxs, N, Q, C);
  }
  maxcoord_kernel<<<1, 256, 0, stream>>>(boxes, N * 4, maxc);
  sort_kernel<<<1, SORT_T, 0, stream>>>(scores, N, sidx, sscore);
  offbox_kernel<<<(N + 255) / 256, 256, 0, stream>>>(sidx, boxes, labels, maxc, N, offb);
  {
    dim3 grid(mwords, (N + NMS_TPB - 1) / NMS_TPB);
    iou_mask_kernel<<<grid, NMS_TPB, 0, stream>>>(offb, N, mwords, mask);
  }
  nms_finalize_kernel<<<1, 32, 0, stream>>>(mask, sscore, sidx, boxes, labels,
                                            pscores, plabels, pboxs,
                                            N, mwords, K, (float*)d_out);
}